// Myloss_82995948028269
// MI455X (gfx1250) — compile-verified
//
#include <hip/hip_runtime.h>

#define FRAMES 8
#define H 128
#define W 128
#define OUT_LEN (H + FRAMES - 1)            /* 135 */
#define NPOS (OUT_LEN * W)                  /* 17280 */
#define NPAIR (NPOS / 2)                    /* 8640  */
#define BLOCK 256
#define ITERS ((NPAIR + BLOCK - 1) / BLOCK) /* 34 */
#define A_COEF 1.0f
#define B_COEF 0.1f

typedef __attribute__((ext_vector_type(2))) float v2f;
typedef __attribute__((ext_vector_type(8))) float v8f;

// One block per sample. Gather-form overlap-add (INTER=1 => o = f + r maps each
// input row to exactly one output row, so no scatter conflicts, each input
// element is read exactly once). The (o-t)^2 sum is accumulated on the matrix
// pipe: wmma_f32_16x16x4_f32 with identical A/B operands accumulates sum of
// squares on the diagonal of C (A*A^T trace identity).
__global__ __launch_bounds__(BLOCK) void tsc_loss_partials(
    const float* __restrict__ outp, const float* __restrict__ tgtp,
    const float* __restrict__ mask, float* __restrict__ partial)
{
    __shared__ float4 tiles[NPAIR];   // (buf_out.x, buf_out.y, buf_tgt.x, buf_tgt.y)
    __shared__ float4 red4[BLOCK];    // reduction scratch

    const int b    = blockIdx.x;
    const int t    = threadIdx.x;
    const int lane = t & 31;          // wave32

    const float* ob = outp + (size_t)b * (FRAMES * H * W);
    const float* tb = tgtp + (size_t)b * (FRAMES * H * W);

    v8f c = {};  // WMMA accumulator: diagonal collects sum((o-t)^2)

    float mn_o =  3.4e38f, mx_o = -3.4e38f;
    float mn_t =  3.4e38f, mx_t = -3.4e38f;

    for (int it = 0; it < ITERS; ++it) {
        const int  pair   = t + it * BLOCK;
        const int  pp     = pair * 2;        // element index into [OUT_LEN x W]
        const bool active = pair < NPAIR;
        const int  o = pp >> 7;              // / W
        const int  w = pp & (W - 1);
        float box = 0.f, boy = 0.f, btx = 0.f, bty = 0.f;
#pragma unroll
        for (int f = 0; f < FRAMES; ++f) {
            const int  r     = o - f;
            const bool valid = active && (r >= 0) && (r < H);
            float xox = 0.f, xoy = 0.f, xtx = 0.f, xty = 0.f, m0 = 0.f, m1 = 0.f;
            if (valid) {
                const size_t off = ((size_t)f * H + r) * W + w;
                float2 xo = *(const float2*)(ob + off);
                float2 xt = *(const float2*)(tb + off);
                float2 mv = *(const float2*)(mask + (size_t)r * W + w);
                xox = xo.x; xoy = xo.y; xtx = xt.x; xty = xt.y; m0 = mv.x; m1 = mv.y;
            }
            // diff for MSE term; zeros for padded/invalid slots contribute 0
            v2f a; a.x = xox - xtx; a.y = xoy - xty;
            // D = A*A^T + C : diagonal accumulates a.x^2 + a.y^2 per lane
            c = __builtin_amdgcn_wmma_f32_16x16x4_f32(
                    false, a, false, a, (short)0, c, false, false);
            // overlap-add contributions (VALU, co-executes with matrix pipe)
            box = fmaf(xox, m0, box); boy = fmaf(xoy, m1, boy);
            btx = fmaf(xtx, m0, btx); bty = fmaf(xty, m1, bty);
        }
        if (active) {
            mn_o = fminf(mn_o, fminf(box, boy));
            mx_o = fmaxf(mx_o, fmaxf(box, boy));
            mn_t = fminf(mn_t, fminf(btx, bty));
            mx_t = fmaxf(mx_t, fmaxf(btx, bty));
            tiles[pair] = make_float4(box, boy, btx, bty);
        }
    }

    // Trace extraction: element (m,n) lives in VGPR m%8; m<8 -> lane n,
    // m>=8 -> lane n+16. Diagonal m==n: VGPR r at lane r and lane r+24.
    float s = 0.f;
#pragma unroll
    for (int rr = 0; rr < 8; ++rr) {
        float cv = c[rr];
        s += ((lane == rr) || (lane == rr + 24)) ? cv : 0.f;
    }

    // Block min/max reduction: components (min_o, min_t, max_o, max_t)
    red4[t] = make_float4(mn_o, mn_t, mx_o, mx_t);
    __syncthreads();
    for (int st = BLOCK / 2; st > 0; st >>= 1) {
        if (t < st) {
            float4 aa = red4[t], bb = red4[t + st];
            red4[t] = make_float4(fminf(aa.x, bb.x), fminf(aa.y, bb.y),
                                  fmaxf(aa.z, bb.z), fmaxf(aa.w, bb.w));
        }
        __syncthreads();
    }
    const float4 mm = red4[0];
    __syncthreads();  // red4 reused below

    const float rng_o = mm.z - mm.x;
    const float rng_t = mm.w - mm.y;
    const bool  nz_o  = (rng_o != 0.f);
    const bool  nz_t  = (rng_t != 0.f);

    float sse = 0.f;
    for (int it = 0; it < ITERS; ++it) {
        const int pair = t + it * BLOCK;
        if (pair < NPAIR) {
            float4 v  = tiles[pair];
            float  n0 = nz_o ? (v.x - mm.x) / rng_o : v.x;
            float  n1 = nz_o ? (v.y - mm.x) / rng_o : v.y;
            float  q0 = nz_t ? (v.z - mm.y) / rng_t : v.z;
            float  q1 = nz_t ? (v.w - mm.y) / rng_t : v.w;
            float  d0 = n0 - q0, d1 = n1 - q1;
            sse = fmaf(d0, d0, sse);
            sse = fmaf(d1, d1, sse);
        }
    }

    // Deterministic block sum of per-sample partial loss
    float* red = (float*)red4;
    red[t] = A_COEF * s + B_COEF * sse;
    __syncthreads();
    for (int st = BLOCK / 2; st > 0; st >>= 1) {
        if (t < st) red[t] += red[t + st];
        __syncthreads();
    }
    if (t == 0) partial[b] = red[0];
}

// Fixed-order final sum over per-sample partials -> scalar loss.
__global__ __launch_bounds__(BLOCK) void reduce_partials(
    const float* __restrict__ partial, float* __restrict__ out, int n)
{
    __shared__ float red[BLOCK];
    const int t = threadIdx.x;
    float s = 0.f;
    for (int i = t; i < n; i += BLOCK) s += partial[i];
    red[t] = s;
    __syncthreads();
    for (int st = BLOCK / 2; st > 0; st >>= 1) {
        if (t < st) red[t] += red[t + st];
        __syncthreads();
    }
    if (t == 0) out[0] = red[0];
}

extern "C" void kernel_launch(void* const* d_in, const int* in_sizes, int n_in,
                              void* d_out, int out_size, void* d_ws, size_t ws_size,
                              hipStream_t stream) {
    (void)n_in; (void)out_size; (void)ws_size;
    const float* outp = (const float*)d_in[0];
    const float* tgtp = (const float*)d_in[1];
    const float* mask = (const float*)d_in[2];
    float* partial = (float*)d_ws;          // B floats of scratch
    const int B = in_sizes[0] / (FRAMES * H * W);  // 512

    tsc_loss_partials<<<B, BLOCK, 0, stream>>>(outp, tgtp, mask, partial);
    reduce_partials<<<1, BLOCK, 0, stream>>>(partial, (float*)d_out, B);
}